// Engram_41721312313466
// MI455X (gfx1250) — compile-verified
//
#include <hip/hip_runtime.h>
#include <hip/hip_bf16.h>
#include <math.h>

// ---------------- problem constants (match reference) ----------------
#define B_      4
#define S_      2048
#define G_      4
#define D_      2048
#define H_      8
#define DHEAD_  64
#define E_      512            // H_*DHEAD_
#define M_      (B_*S_)        // 8192 rows
#define NKV_    (G_*D_ + D_)   // 10240 : [key g0..g3 | value]
#define EPS_    1e-6f

typedef float v2f __attribute__((ext_vector_type(2)));
typedef float v8f __attribute__((ext_vector_type(8)));

// ---------------- kernel 1: hashed embedding gather ----------------
// emb[m, h*64 + j] = table[(ids[m*8+h] + offsets[h]) * 64 + j]
__global__ __launch_bounds__(128) void k_gather(const int* __restrict__ ids,
                                                const int* __restrict__ offsets,
                                                const float* __restrict__ table,
                                                float* __restrict__ emb) {
  const int m = blockIdx.x;              // 0..8191
  const int t = threadIdx.x;             // 128 threads, one float4 each
  const int h = t >> 4;                  // 16 threads per head
  const int j = (t & 15) << 2;           // 0..60
  const long long row = (long long)ids[m * H_ + h] + offsets[h];
  const float4 v = *(const float4*)(table + row * DHEAD_ + j);
  *(float4*)(emb + (size_t)m * E_ + h * DHEAD_ + j) = v;
}

// ---------------- kernel 2: fused fp32 WMMA GEMM ----------------
// C[8192, 10240] = emb[8192,512] @ [Wk(g=0..3) | Wv]  (all K-major, ldb = 2048)
// A tile kept row-major in LDS; B tile stored TRANSPOSED ([n][k]) so both A and
// B fragments are single contiguous ds_load_b64 (even-aligned pairs, no repack).
#define BM 128
#define BN 128
#define BK 32
#define LDA_S 34    // padded LDS strides (34 mod 64 -> conflict-free for 16 rows)
#define LDBT_S 34

__global__ __launch_bounds__(256) void k_gemm_kv(const float* __restrict__ A,
                                                 const float* __restrict__ Wk,
                                                 const float* __restrict__ Wv,
                                                 float* __restrict__ C) {
  __shared__ float sA[BM * LDA_S];     // [m][k]
  __shared__ float sBt[BN * LDBT_S];   // [n][k]  (transposed)

  const int tid  = threadIdx.x;
  const int lane = tid & 31;
  const int wave = tid >> 5;
  const int hi   = lane >> 4;      // 0: lanes 0-15, 1: lanes 16-31
  const int l16  = lane & 15;

  const int m0 = blockIdx.y * BM;
  const int n0 = blockIdx.x * BN;

  // resolve which weight segment this N-tile lives in (never crosses: 128 | 2048)
  const float* Bbase;
  int ncol;
  if (n0 < G_ * D_) { const int g = n0 >> 11; Bbase = Wk + (size_t)g * E_ * D_; ncol = n0 & (D_ - 1); }
  else              { Bbase = Wv; ncol = n0 - G_ * D_; }

  // 8 waves as 4(M) x 2(N); each wave owns a 32x64 tile = 2x4 WMMA 16x16 tiles
  const int wm = (wave & 3) * 32;
  const int wn = (wave >> 2) * 64;

  v8f acc[2][4];
#pragma unroll
  for (int mi = 0; mi < 2; ++mi)
#pragma unroll
    for (int ni = 0; ni < 4; ++ni)
      acc[mi][ni] = (v8f){0.f, 0.f, 0.f, 0.f, 0.f, 0.f, 0.f, 0.f};

  // staging decomposition for A: 1024 float4 units; for B: 1024 k-quad units
  const int a_r = (tid * 4) >> 5;            // reused below via idx math
  (void)a_r;

  for (int kb = 0; kb < E_; kb += BK) {
    // ---- stage A tile 128x32 (float4 loads, 4 per thread) ----
#pragma unroll
    for (int i = 0; i < 4; ++i) {
      const int idx = tid + i * 256;
      const int r = idx >> 3;                 // row m (0..127)
      const int c = (idx & 7) << 2;           // col k (0..28)
      const float4 v = *(const float4*)(A + (size_t)(m0 + r) * E_ + kb + c);
      float* dst = sA + r * LDA_S + c;
      dst[0] = v.x; dst[1] = v.y; dst[2] = v.z; dst[3] = v.w;
    }
    // ---- stage B tile 32x128, transposed into sBt[n][k] ----
    // unit u: n = u&127 (lane-consecutive -> coalesced), kq = (u>>7)*4
#pragma unroll
    for (int i = 0; i < 4; ++i) {
      const int u = tid + i * 256;
      const int n = u & 127;
      const int kq = (u >> 7) << 2;
      const float* gsrc = Bbase + (size_t)(kb + kq) * D_ + ncol + n;
      float* dst = sBt + n * LDBT_S + kq;     // 8B-aligned (34n even, kq%4==0)
      dst[0] = gsrc[0];
      dst[1] = gsrc[D_];
      dst[2] = gsrc[2 * D_];
      dst[3] = gsrc[3 * D_];
    }
    // prefetch next K-tiles into cache while this tile computes
    if (kb + BK < E_) {
      const int pr = tid >> 1;                // 128 rows, 2 threads/row
      __builtin_prefetch(A + (size_t)(m0 + pr) * E_ + kb + BK + (tid & 1) * 16, 0, 3);
      const int pk = tid & 31;
      const int pn = (tid >> 5) * 32;
      __builtin_prefetch(Bbase + (size_t)(kb + BK + pk) * D_ + ncol + pn, 0, 3);
    }
    __syncthreads();

#pragma unroll
    for (int kk = 0; kk < BK; kk += 4) {
      // A frag (16x4 f32): vgpr0 = A[m][kk+2*hi], vgpr1 = A[m][kk+2*hi+1]
      v2f af[2];
#pragma unroll
      for (int mi = 0; mi < 2; ++mi) {
        const float* p = sA + (wm + mi * 16 + l16) * LDA_S + kk + 2 * hi;
        af[mi].x = p[0];
        af[mi].y = p[1];
      }
      // B frag (4x16 f32): vgpr0 = B[kk+2*hi][n], vgpr1 = B[kk+2*hi+1][n]
      // transposed layout -> contiguous pair at sBt[n][kk+2*hi]
      v2f bf[4];
#pragma unroll
      for (int ni = 0; ni < 4; ++ni) {
        const float* p = sBt + (wn + ni * 16 + l16) * LDBT_S + kk + 2 * hi;
        bf[ni].x = p[0];
        bf[ni].y = p[1];
      }
#pragma unroll
      for (int mi = 0; mi < 2; ++mi)
#pragma unroll
        for (int ni = 0; ni < 4; ++ni)
          acc[mi][ni] = __builtin_amdgcn_wmma_f32_16x16x4_f32(
              false, af[mi], false, bf[ni], (short)0, acc[mi][ni], false, false);
    }
    __syncthreads();
  }

  // C layout: vgpr r -> M = r (lanes 0-15) / 8+r (lanes 16-31), N = lane&15
#pragma unroll
  for (int mi = 0; mi < 2; ++mi) {
#pragma unroll
    for (int ni = 0; ni < 4; ++ni) {
      const int rbase = m0 + wm + mi * 16 + hi * 8;
      const int cg = n0 + wn + ni * 16 + l16;
#pragma unroll
      for (int r = 0; r < 8; ++r)
        C[(size_t)(rbase + r) * NKV_ + cg] = acc[mi][ni][r];
    }
  }
}

// ---------------- block reduction helper (wave32) ----------------
__device__ __forceinline__ float block_sum(float v, float* sbuf) {
#pragma unroll
  for (int off = 16; off; off >>= 1) v += __shfl_xor(v, off, 32);
  const int wid = threadIdx.x >> 5;
  if ((threadIdx.x & 31) == 0) sbuf[wid] = v;
  __syncthreads();
  float r = (threadIdx.x < 8) ? sbuf[threadIdx.x] : 0.f;
  if (threadIdx.x < 32) {
#pragma unroll
    for (int off = 4; off; off >>= 1) r += __shfl_xor(r, off, 32);
    if (threadIdx.x == 0) sbuf[0] = r;
  }
  __syncthreads();
  const float res = sbuf[0];
  __syncthreads();
  return res;
}

// ---------------- kernel 3: RMSNorms + gate + value gating + conv-input norm ----
// one block per (b,s,g) row of D=2048; 256 threads x 8 elems
__global__ __launch_bounds__(256) void k_gate(const float* __restrict__ hidden,
                                              const float* __restrict__ kv,
                                              const float* __restrict__ bk,
                                              const float* __restrict__ kscale,
                                              const float* __restrict__ qscale,
                                              const float* __restrict__ bv,
                                              const float* __restrict__ cscale,
                                              float* __restrict__ out,
                                              float* __restrict__ xn) {
  __shared__ float sred[8];
  const int row = blockIdx.x;          // ((b*S+s)*G + g)
  const int g = row & (G_ - 1);
  const int m = row >> 2;
  const float* hrow = hidden + (size_t)row * D_;
  const float* krow = kv + (size_t)m * NKV_ + (size_t)g * D_;
  const float* vrow = kv + (size_t)m * NKV_ + (size_t)(G_ * D_);

  float h[8], kr[8];
  float s_h2 = 0.f, s_k2 = 0.f, s_dot = 0.f;
#pragma unroll
  for (int i = 0; i < 8; ++i) {
    const int d = threadIdx.x + i * 256;
    h[i] = hrow[d];
    kr[i] = krow[d] + bk[g * D_ + d];
    const float qk = qscale[g * D_ + d] * kscale[g * D_ + d];
    s_h2 += h[i] * h[i];
    s_k2 += kr[i] * kr[i];
    s_dot += h[i] * kr[i] * qk;
  }
  s_h2 = block_sum(s_h2, sred);
  s_k2 = block_sum(s_k2, sred);
  s_dot = block_sum(s_dot, sred);

  const float rinv_h = rsqrtf(s_h2 * (1.0f / D_) + EPS_);
  const float rinv_k = rsqrtf(s_k2 * (1.0f / D_) + EPS_);
  const float gd = s_dot * rinv_h * rinv_k * 0.022097086912079608f; // 1/sqrt(2048)
  const float sq = sqrtf(fmaxf(fabsf(gd), 1e-6f));
  const float arg = (gd > 0.f) ? sq : ((gd < 0.f) ? -sq : 0.f);
  const float gate = 1.0f / (1.0f + __expf(-arg));

  float gv[8];
  float s_g2 = 0.f;
#pragma unroll
  for (int i = 0; i < 8; ++i) {
    const int d = threadIdx.x + i * 256;
    const float v = vrow[d] + bv[d];
    gv[i] = gate * v;
    out[(size_t)row * D_ + d] = gv[i];
    s_g2 += gv[i] * gv[i];
  }
  s_g2 = block_sum(s_g2, sred);
  const float rinv_g = rsqrtf(s_g2 * (1.0f / D_) + EPS_);
#pragma unroll
  for (int i = 0; i < 8; ++i) {
    const int d = threadIdx.x + i * 256;
    xn[(size_t)row * D_ + d] = gv[i] * rinv_g * cscale[g * D_ + d];
  }
}

// ---------------- kernel 4: causal dilated depthwise conv + SiLU + residual ----
// out[b,s,c] += silu( sum_k cw[k,c] * xn[b, s-9+3k, c] ),  c over G*D = 8192
__global__ __launch_bounds__(256) void k_conv(const float* __restrict__ xn,
                                              const float* __restrict__ cw,
                                              float* __restrict__ out) {
  const size_t idx4 = (size_t)blockIdx.x * blockDim.x + threadIdx.x; // float4 id
  const int c4 = (int)(idx4 & (2048 - 1));          // 8192/4 = 2048 float4 per row
  const size_t bs = idx4 >> 11;                     // b*S + s
  const int s = (int)(bs & (S_ - 1));
  const int c = c4 << 2;

  float4 y = make_float4(0.f, 0.f, 0.f, 0.f);
#pragma unroll
  for (int k = 0; k < 4; ++k) {
    const int ss = s - 9 + 3 * k;
    if (ss >= 0) {
      const float4 xv = *(const float4*)(xn + (bs - 9 + 3 * (size_t)k) * (G_ * D_) + c);
      const float4 wv = *(const float4*)(cw + (size_t)k * (G_ * D_) + c);
      y.x += xv.x * wv.x; y.y += xv.y * wv.y;
      y.z += xv.z * wv.z; y.w += xv.w * wv.w;
    }
  }
  float4* po = (float4*)(out + bs * (G_ * D_) + c);
  float4 o = *po;
  o.x += y.x / (1.0f + __expf(-y.x));
  o.y += y.y / (1.0f + __expf(-y.y));
  o.z += y.z / (1.0f + __expf(-y.z));
  o.w += y.w / (1.0f + __expf(-y.w));
  *po = o;
}

// ---------------- host launcher ----------------
extern "C" void kernel_launch(void* const* d_in, const int* in_sizes, int n_in,
                              void* d_out, int out_size, void* d_ws, size_t ws_size,
                              hipStream_t stream) {
  const float* hidden  = (const float*)d_in[0];   // (B,S,G,D)
  const int*   ids     = (const int*)  d_in[1];   // (B,S,H)
  const int*   offsets = (const int*)  d_in[2];   // (H,)
  const float* table   = (const float*)d_in[3];   // (Vtot, 64)
  const float* Wk      = (const float*)d_in[4];   // (G,E,D)
  const float* bk      = (const float*)d_in[5];   // (G,D)
  const float* kscale  = (const float*)d_in[6];   // (G,D)
  const float* qscale  = (const float*)d_in[7];   // (G,D)
  const float* Wv      = (const float*)d_in[8];   // (E,D)
  const float* bv      = (const float*)d_in[9];   // (D,)
  const float* cscale  = (const float*)d_in[10];  // (G,D)
  const float* cw      = (const float*)d_in[11];  // (K, G*D)
  float* out = (float*)d_out;                     // (B,S,G,D)

  // workspace carve-out: emb | kv | xn
  char* ws = (char*)d_ws;
  float* emb = (float*)ws;                                       // 8192*512
  float* kv  = (float*)(ws + (size_t)M_ * E_ * sizeof(float));   // 8192*10240
  float* xn  = (float*)(ws + (size_t)M_ * E_ * sizeof(float)
                           + (size_t)M_ * NKV_ * sizeof(float)); // 4*2048*8192

  // 1) gather embeddings
  k_gather<<<M_, 128, 0, stream>>>(ids, offsets, table, emb);

  // 2) fused WMMA GEMM: kv = emb @ [Wk|Wv]
  dim3 gg(NKV_ / BN, M_ / BM);     // 80 x 64
  k_gemm_kv<<<gg, 256, 0, stream>>>(emb, Wk, Wv, kv);

  // 3) gate fusion: writes gated_value -> out, normalized conv input -> xn
  k_gate<<<M_ * G_, 256, 0, stream>>>(hidden, kv, bk, kscale, qscale, bv,
                                      cscale, out, xn);

  // 4) causal dilated depthwise conv + SiLU + residual (in-place on out)
  const size_t n4 = (size_t)B_ * S_ * G_ * D_ / 4;  // 16,777,216 float4s
  k_conv<<<(unsigned)(n4 / 256), 256, 0, stream>>>(xn, cw, out);
}